// GATAnomaly_62809601736708
// MI455X (gfx1250) — compile-verified
//
#include <hip/hip_runtime.h>
#include <math.h>

typedef __attribute__((ext_vector_type(2))) float v2f;
typedef __attribute__((ext_vector_type(8))) float v8f;

// ---------------------------------------------------------------------------
// Dense GEMM  D[M,Ncols] = A[M,K] x B[K,Ncols]  with V_WMMA_F32_16X16X4_F32.
// Block = 128 threads = 4 waves; block computes a 16(M) x 64(N) tile.
// K is processed in 32-wide chunks staged to LDS. Full chunks stage with
// unguarded, batched loads (all global loads issued before any LDS store so
// they pipeline); only the K-tail chunk takes the guarded zero-pad path.
// The WMMA loop has no predication (EXEC all-ones) and uses two accumulator
// chains to break the C-matrix RAW dependency.
// Fragment layouts per CDNA5 ISA 7.12.2:
//   A 16x4 : lane holds row M=lane&15; element i has K = 2*(lane>>4) + i
//   B 4x16 : lane holds col N=lane&15; element i has K = 2*(lane>>4) + i
//   C 16x16: VGPR r -> M = r + 8*(lane>>4), N = lane&15
// ---------------------------------------------------------------------------
#define TK 32

__global__ void gat_gemm_wmma(const float* __restrict__ A, const float* __restrict__ B,
                              float* __restrict__ D, int K, int Ncols)
{
    __shared__ float Alds[16 * TK];      // [m][k], stride TK
    __shared__ float Blds[TK * 64];      // [k][colLocal], stride 64

    const int t    = threadIdx.x;        // 0..127
    const int lane = t & 31;
    const int wave = t >> 5;
    const int mt   = blockIdx.x;
    const int m    = lane & 15;
    const int kb   = (lane >> 4) << 1;   // 0 or 2
    const int colL = wave * 16 + m;      // 0..63 within block tile
    const int col0 = blockIdx.y * 64;

    v8f c0 = {}, c1 = {};
    const int nchunks = (K + TK - 1) / TK;
    for (int ch = 0; ch < nchunks; ++ch) {
        const int kbase = ch * TK;
        if (kbase + TK <= K) {
            // ---------- full chunk: batched unguarded staging ----------
            float  va[4];
            float4 vb[4];
#pragma unroll
            for (int j = 0; j < 4; ++j) {            // A: 16x32 scalars
                const int idx = t + j * 128;
                const int r = idx >> 5, k = idx & (TK - 1);
                va[j] = A[(size_t)(mt * 16 + r) * K + kbase + k];
            }
#pragma unroll
            for (int j = 0; j < 4; ++j) {            // B: 32x64 as float4
                const int q = t + j * 128;
                const int k = q >> 4, c4 = (q & 15) << 2;
                vb[j] = *(const float4*)&B[(size_t)(kbase + k) * Ncols + col0 + c4];
            }
#pragma unroll
            for (int j = 0; j < 4; ++j)
                Alds[t + j * 128] = va[j];
#pragma unroll
            for (int j = 0; j < 4; ++j) {
                const int q = t + j * 128;
                const int k = q >> 4, c4 = (q & 15) << 2;
                *(float4*)&Blds[(k << 6) + c4] = vb[j];
            }
        } else {
            // ---------- K-tail chunk: guarded, zero-padded ----------
#pragma unroll
            for (int idx = t; idx < 16 * TK; idx += 128) {
                const int r = idx >> 5, k = idx & (TK - 1);
                const int kg = kbase + k;
                Alds[idx] = (kg < K) ? A[(size_t)(mt * 16 + r) * K + kg] : 0.f;
            }
#pragma unroll
            for (int idx = t; idx < TK * 64; idx += 128) {
                const int k = idx >> 6, cc = idx & 63;
                const int kg = kbase + k;
                Blds[idx] = (kg < K) ? B[(size_t)kg * Ncols + col0 + cc] : 0.f;
            }
        }
        __syncthreads();
        // prefetch next B chunk into cache while this chunk computes
        if (ch + 1 < nchunks)
            __builtin_prefetch(&B[(size_t)(kbase + TK) * Ncols + col0 + colL], 0, 1);
        // ---- 8 unpredicated WMMA steps, two independent accumulator chains
#pragma unroll
        for (int kk = 0; kk < TK; kk += 8) {
            const v2f a0 = *(const v2f*)&Alds[m * TK + kk + kb];       // 8B aligned
            v2f b0;
            b0.x = Blds[(kk + kb) * 64 + colL];
            b0.y = Blds[(kk + kb + 1) * 64 + colL];
            c0 = __builtin_amdgcn_wmma_f32_16x16x4_f32(false, a0, false, b0,
                                                       (short)0, c0, false, false);
            const v2f a1 = *(const v2f*)&Alds[m * TK + kk + 4 + kb];
            v2f b1;
            b1.x = Blds[(kk + 4 + kb) * 64 + colL];
            b1.y = Blds[(kk + 5 + kb) * 64 + colL];
            c1 = __builtin_amdgcn_wmma_f32_16x16x4_f32(false, a1, false, b1,
                                                       (short)0, c1, false, false);
        }
        __syncthreads();
    }
    const int mbase = mt * 16 + ((lane >> 4) << 3);
    const int col   = col0 + colL;
#pragma unroll
    for (int r = 0; r < 8; ++r)
        D[(size_t)(mbase + r) * Ncols + col] = c0[r] + c1[r];
}

// ---------------------------------------------------------------------------
// Helpers
// ---------------------------------------------------------------------------
__device__ __forceinline__ void edge_ep(const int* __restrict__ ei, int e, int E,
                                        int& s, int& d)
{
    if (e < E) { s = ei[e]; d = ei[E + e]; }      // edge_index rows 0 / 1
    else       { s = d = e - E; }                 // appended self loop
}

__device__ __forceinline__ void atomic_max_f32(float* addr, float v)
{
    // order-preserving int mapping; location init to -inf (bits 0xFF800000)
    if (v >= 0.f) atomicMax((int*)addr,          __float_as_int(v));
    else          atomicMin((unsigned int*)addr, __float_as_uint(v));
}

__global__ void gat_fill(float* __restrict__ p, float v, int n)
{
    int i = blockIdx.x * blockDim.x + threadIdx.x;
    if (i < n) p[i] = v;
}

// es[n,h] = <h[n,h,:], a_src[h,:]> ; ed likewise  (float4 path when C%4==0)
__global__ void gat_attn_coef(const float* __restrict__ h, const float* __restrict__ asrc,
                              const float* __restrict__ adst, float* __restrict__ es,
                              float* __restrict__ ed, int n, int H, int C)
{
    int i = blockIdx.x * blockDim.x + threadIdx.x;
    if (i >= n * H) return;
    int node = i / H, head = i - node * H;
    const float* hp = h + (size_t)node * H * C + (size_t)head * C;
    const float* as = asrc + head * C;
    const float* ad = adst + head * C;
    float s = 0.f, d = 0.f;
    if ((C & 3) == 0) {
        const float4* hp4 = (const float4*)hp;
        const float4* as4 = (const float4*)as;
        const float4* ad4 = (const float4*)ad;
        for (int c = 0; c < (C >> 2); ++c) {
            float4 v = hp4[c], a = as4[c], b = ad4[c];
            s += v.x * a.x + v.y * a.y + v.z * a.z + v.w * a.w;
            d += v.x * b.x + v.y * b.y + v.z * b.z + v.w * b.w;
        }
    } else {
        for (int c = 0; c < C; ++c) { float v = hp[c]; s += v * as[c]; d += v * ad[c]; }
    }
    es[i] = s; ed[i] = d;
}

// pass 1: raw score + leaky relu, atomic max per (dst, head)
__global__ void gat_edge_score(const int* __restrict__ ei, const float* __restrict__ es,
                               const float* __restrict__ ed, float* __restrict__ sc,
                               float* __restrict__ m, int E, int E2n, int H)
{
    int i = blockIdx.x * blockDim.x + threadIdx.x;
    if (i >= E2n * H) return;
    int e = i / H, h = i - e * H;
    int s, d; edge_ep(ei, e, E, s, d);
    float v = es[s * H + h] + ed[d * H + h];
    v = (v > 0.f) ? v : 0.2f * v;                 // leaky_relu 0.2
    sc[i] = v;
    atomic_max_f32(&m[d * H + h], v);
}

// pass 2: p = exp(e - m[dst]); atomic sum per (dst, head)
__global__ void gat_edge_exp(const int* __restrict__ ei, float* __restrict__ sc,
                             const float* __restrict__ m, float* __restrict__ ssum,
                             int E, int E2n, int H)
{
    int i = blockIdx.x * blockDim.x + threadIdx.x;
    if (i >= E2n * H) return;
    int e = i / H, h = i - e * H;
    int s, d; edge_ep(ei, e, E, s, d);
    (void)s;
    float p = expf(sc[i] - m[d * H + h]);
    sc[i] = p;
    atomicAdd(&ssum[d * H + h], p);
}

// pass 3: agg[dst] += h[src] * alpha ; one block per edge, lanes over channels
__global__ void gat_edge_agg(const int* __restrict__ ei, const float* __restrict__ h,
                             const float* __restrict__ sc, const float* __restrict__ ssum,
                             float* __restrict__ agg, int E, int E2n, int H, int C)
{
    int e = blockIdx.x;
    if (e >= E2n) return;
    int s, d; edge_ep(ei, e, E, s, d);
    __shared__ float alph[8];
    if ((int)threadIdx.x < H)
        alph[threadIdx.x] = sc[e * H + threadIdx.x] / ssum[d * H + threadIdx.x];
    __syncthreads();
    const int HC = H * C;
    for (int c = threadIdx.x; c < HC; c += blockDim.x)
        atomicAdd(&agg[(size_t)d * HC + c], h[(size_t)s * HC + c] * alph[c / C]);
}

__global__ void gat_bias_elu(float* __restrict__ a, const float* __restrict__ b,
                             int total, int HC, int do_elu)
{
    int i = blockIdx.x * blockDim.x + threadIdx.x;
    if (i >= total) return;
    float v = a[i] + b[i % HC];
    if (do_elu) v = (v > 0.f) ? v : (expf(v) - 1.f);
    a[i] = v;
}

// layer-3 projection: o[n] = <h[n,0:64], W3[:,0]>
__global__ void gat_gemv64(const float* __restrict__ h, const float* __restrict__ w3,
                           float* __restrict__ o, int n)
{
    int i = blockIdx.x * blockDim.x + threadIdx.x;
    if (i >= n) return;
    const float4* hp = (const float4*)(h + (size_t)i * 64);
    const float4* wp = (const float4*)w3;
    float acc = 0.f;
#pragma unroll
    for (int c = 0; c < 16; ++c) {
        float4 v = hp[c], w = wp[c];
        acc += v.x * w.x + v.y * w.y + v.z * w.z + v.w * w.w;
    }
    o[i] = acc;
}

// out[0] = mean(agg) + b3[0]
__global__ void gat_finalize(const float* __restrict__ agg, const float* __restrict__ b3,
                             float* __restrict__ out, int n)
{
    __shared__ float sm[256];
    float acc = 0.f;
    for (int i = blockIdx.x * blockDim.x + threadIdx.x; i < n; i += gridDim.x * blockDim.x)
        acc += agg[i];
    sm[threadIdx.x] = acc;
    __syncthreads();
    for (int s = 128; s > 0; s >>= 1) {
        if ((int)threadIdx.x < s) sm[threadIdx.x] += sm[threadIdx.x + s];
        __syncthreads();
    }
    if (threadIdx.x == 0) atomicAdd(out, sm[0] / (float)n);
    if (blockIdx.x == 0 && threadIdx.x == 0) atomicAdd(out, b3[0]);
}

// ---------------------------------------------------------------------------
extern "C" void kernel_launch(void* const* d_in, const int* in_sizes, int n_in,
                              void* d_out, int out_size, void* d_ws, size_t ws_size,
                              hipStream_t stream)
{
    (void)in_sizes; (void)n_in; (void)out_size; (void)ws_size;
    const float* x   = (const float*)d_in[0];
    const float* W1  = (const float*)d_in[1];
    const float* as1 = (const float*)d_in[2];
    const float* ad1 = (const float*)d_in[3];
    const float* b1  = (const float*)d_in[4];
    const float* W2  = (const float*)d_in[5];
    const float* as2 = (const float*)d_in[6];
    const float* ad2 = (const float*)d_in[7];
    const float* b2  = (const float*)d_in[8];
    const float* W3  = (const float*)d_in[9];
    const float* as3 = (const float*)d_in[10];
    const float* ad3 = (const float*)d_in[11];
    const float* b3  = (const float*)d_in[12];
    const int*   ei  = (const int*)d_in[13];

    const int N = 50000, E = 400000, EE = N + E;   // EE = edges + self loops
    const float NEG_INF = -INFINITY;

    // ---- workspace layout (floats) ---------------------------------------
    float* w     = (float*)d_ws;
    float* h1pre = w;                               // N*512
    float* agg1  = h1pre + (size_t)N * 512;         // N*512 (-> h1 activated)
    float* p1    = agg1  + (size_t)N * 512;         // EE*4
    float* es1   = p1 + (size_t)EE * 4;             // N*4
    float* ed1   = es1 + (size_t)N * 4;
    float* m1    = ed1 + (size_t)N * 4;
    float* s1    = m1  + (size_t)N * 4;
    // layer 2/3 buffers reuse the h1pre panel (free after layer-1 agg pass)
    float* h2pre = h1pre;                           // N*64
    float* agg2  = h2pre + (size_t)N * 64;          // N*64 (-> h2 activated)
    float* p2    = agg2  + (size_t)N * 64;          // EE
    float* es2   = p2 + EE;  float* ed2 = es2 + N;
    float* m2    = ed2 + N;  float* s2  = m2 + N;
    float* h3pre = s2 + N;                          // N
    float* p3    = h3pre + N;                       // EE
    float* es3   = p3 + EE;  float* ed3 = es3 + N;
    float* m3    = ed3 + N;  float* s3  = m3 + N;
    float* agg3  = s3 + N;                          // N

    // ======================= layer 1 (H=4, C=128) =========================
    gat_gemm_wmma<<<dim3(N / 16, 8), 128, 0, stream>>>(x, W1, h1pre, 21, 512);
    gat_attn_coef<<<(N * 4 + 255) / 256, 256, 0, stream>>>(h1pre, as1, ad1, es1, ed1, N, 4, 128);
    gat_fill<<<(N * 4 + 255) / 256, 256, 0, stream>>>(m1, NEG_INF, N * 4);
    hipMemsetAsync(s1,   0, (size_t)N * 4   * sizeof(float), stream);
    hipMemsetAsync(agg1, 0, (size_t)N * 512 * sizeof(float), stream);
    gat_edge_score<<<(EE * 4 + 255) / 256, 256, 0, stream>>>(ei, es1, ed1, p1, m1, E, EE, 4);
    gat_edge_exp  <<<(EE * 4 + 255) / 256, 256, 0, stream>>>(ei, p1, m1, s1, E, EE, 4);
    gat_edge_agg  <<<EE, 256, 0, stream>>>(ei, h1pre, p1, s1, agg1, E, EE, 4, 128);
    gat_bias_elu  <<<((size_t)N * 512 + 255) / 256, 256, 0, stream>>>(agg1, b1, N * 512, 512, 1);

    // ======================= layer 2 (H=1, C=64) ==========================
    gat_gemm_wmma<<<dim3(N / 16, 1), 128, 0, stream>>>(agg1, W2, h2pre, 512, 64);
    gat_attn_coef<<<(N + 255) / 256, 256, 0, stream>>>(h2pre, as2, ad2, es2, ed2, N, 1, 64);
    gat_fill<<<(N + 255) / 256, 256, 0, stream>>>(m2, NEG_INF, N);
    hipMemsetAsync(s2,   0, (size_t)N      * sizeof(float), stream);
    hipMemsetAsync(agg2, 0, (size_t)N * 64 * sizeof(float), stream);
    gat_edge_score<<<(EE + 255) / 256, 256, 0, stream>>>(ei, es2, ed2, p2, m2, E, EE, 1);
    gat_edge_exp  <<<(EE + 255) / 256, 256, 0, stream>>>(ei, p2, m2, s2, E, EE, 1);
    gat_edge_agg  <<<EE, 64, 0, stream>>>(ei, h2pre, p2, s2, agg2, E, EE, 1, 64);
    gat_bias_elu  <<<((size_t)N * 64 + 255) / 256, 256, 0, stream>>>(agg2, b2, N * 64, 64, 1);

    // ======================= layer 3 (H=1, C=1) ===========================
    gat_gemv64<<<(N + 255) / 256, 256, 0, stream>>>(agg2, W3, h3pre, N);
    gat_attn_coef<<<(N + 255) / 256, 256, 0, stream>>>(h3pre, as3, ad3, es3, ed3, N, 1, 1);
    gat_fill<<<(N + 255) / 256, 256, 0, stream>>>(m3, NEG_INF, N);
    hipMemsetAsync(s3,   0, (size_t)N * sizeof(float), stream);
    hipMemsetAsync(agg3, 0, (size_t)N * sizeof(float), stream);
    gat_edge_score<<<(EE + 255) / 256, 256, 0, stream>>>(ei, es3, ed3, p3, m3, E, EE, 1);
    gat_edge_exp  <<<(EE + 255) / 256, 256, 0, stream>>>(ei, p3, m3, s3, E, EE, 1);
    gat_edge_agg  <<<EE, 32, 0, stream>>>(ei, h3pre, p3, s3, agg3, E, EE, 1, 1);

    hipMemsetAsync(d_out, 0, sizeof(float), stream);
    gat_finalize<<<256, 256, 0, stream>>>(agg3, b3, (float*)d_out, N);
}